// lstm_6425271075258
// MI455X (gfx1250) — compile-verified
//
#include <hip/hip_runtime.h>

// ---------------------------------------------------------------------------
// 2-layer LSTM scan (S=2048 sequential, B=4096 parallel, H=4, IN=OUT=1),
// fully fused in registers, V_WMMA_F32_16X16X4_F32 for every matmul.
//
// Gate-row permutation trick: A-operand rows loaded as perm[m]=(m%4)*4+m/4
// (i0,f0,g0,o0, i1,f1,g1,o1 | i2..o3), so with the 16x16 C/D layout each lane
// holds the full {i,f,g,o} quartet for TWO hidden units of its batch element,
// and the updated h pair {h[klo],h[klo+1]} is ALREADY the 16x4 B-operand
// layout for the next WMMA.  Steady-state recurrence: zero cross-lane ops,
// zero lane-selects.  Only the OUT=1 projection needs one ds_swizzle/step.
//
// Activations use gfx1250's hardware v_tanh_f32 when available:
//   tanh(x)    -> 1 trans op
//   sigmoid(x) = 0.5*tanh(x/2)+0.5 -> mul + tanh + fma (1 trans op)
// halving transcendental pressure on the latency-bound chain.
//
// 128 waves x 2 independent chains x 16 batch = B = 4096; two chains per wave
// provide ILP to fill the serial-dependency stall slots.
// ---------------------------------------------------------------------------

#define S_LEN 2048
#define B_LEN 4096
#define NCH   2  // independent chains (batch tiles) per wave

typedef __attribute__((ext_vector_type(2))) float v2f;
typedef __attribute__((ext_vector_type(8))) float v8f;

__device__ __forceinline__ float ftanh(float x) {
#if __has_builtin(__builtin_amdgcn_tanhf)
  return __builtin_amdgcn_tanhf(x);  // v_tanh_f32
#else
  return 2.0f * __builtin_amdgcn_rcpf(1.0f + __expf(-2.0f * x)) - 1.0f;
#endif
}
__device__ __forceinline__ float fsig(float x) {
#if __has_builtin(__builtin_amdgcn_tanhf)
  return fmaf(0.5f, __builtin_amdgcn_tanhf(0.5f * x), 0.5f);
#else
  return __builtin_amdgcn_rcpf(1.0f + __expf(-x));
#endif
}

// One hidden unit's LSTM cell update from its {i,f,g,o} quartet.
__device__ __forceinline__ void cell4(float di, float df, float dg, float do_,
                                      float& c, float& h) {
  float ig = fsig(di);
  float fg = fsig(df);
  float gg = ftanh(dg);
  float og = fsig(do_);
  c = fg * c + ig * gg;
  h = og * ftanh(c);
}

__global__ __launch_bounds__(32) void lstm2_wmma_kernel(
    const float* __restrict__ x,
    const float* __restrict__ W_ih0, const float* __restrict__ W_hh0,
    const float* __restrict__ b_ih0, const float* __restrict__ b_hh0,
    const float* __restrict__ W_ih1, const float* __restrict__ W_hh1,
    const float* __restrict__ b_ih1, const float* __restrict__ b_hh1,
    const float* __restrict__ W_out, const float* __restrict__ b_out,
    float* __restrict__ out) {
  const int  lane = threadIdx.x;  // 0..31, wave32
  const int  n    = lane & 15;    // batch-in-tile = N index = A/B "row"
  const bool hi   = lane >= 16;
  const int  klo  = hi ? 2 : 0;   // K offset in A/B layout == first hidden unit

  // ---- A operands with permuted gate rows: A[m][k] = W[perm[m]][k], m=n ----
  const int pm = (n & 3) * 4 + (n >> 2);  // perm[n]
  v2f A0, A1i, A1h;
  A0.x  = W_hh0[pm * 4 + klo];  A0.y  = W_hh0[pm * 4 + klo + 1];
  A1i.x = W_ih1[pm * 4 + klo];  A1i.y = W_ih1[pm * 4 + klo + 1];
  A1h.x = W_hh1[pm * 4 + klo];  A1h.y = W_hh1[pm * 4 + klo + 1];

  // ---- per-lane C/D row constants: WMMA row m_r = (hi?8:0)+r, permuted ----
  float w0g[8], bs0[8];
  v8f   bias1;
#pragma unroll
  for (int r = 0; r < 8; ++r) {
    int m    = (hi ? 8 : 0) + r;
    int g    = (m & 3) * 4 + (m >> 2);  // original PyTorch gate row
    w0g[r]   = W_ih0[g];                // (16,1) input weight column
    bs0[r]   = b_ih0[g] + b_hh0[g];
    bias1[r] = b_ih1[g] + b_hh1[g];
  }
  // output projection: this lane owns hidden units klo, klo+1
  const float woA = W_out[klo], woB = W_out[klo + 1];
  const float bo  = b_out[0];

  // ---- per-chain state: h/c pair for this lane's two hidden units ---------
  float h0a[NCH], h0b[NCH], c0a[NCH], c0b[NCH];
  float h1a[NCH], h1b[NCH], c1a[NCH], c1b[NCH];
  const float* xp[NCH];
  float*       op[NCH];
  float        xv[NCH];
#pragma unroll
  for (int ch = 0; ch < NCH; ++ch) {
    h0a[ch] = h0b[ch] = c0a[ch] = c0b[ch] = 0.0f;
    h1a[ch] = h1b[ch] = c1a[ch] = c1b[ch] = 0.0f;
    int b  = (blockIdx.x * NCH + ch) * 16 + n;
    xp[ch] = x + b;
    op[ch] = out + b;
    xv[ch] = xp[ch][0];  // software-pipelined x value for step 0
  }

  for (int s = 0; s < S_LEN; ++s) {
    // prefetch next step's x (wrap-safe on the final iteration)
    const size_t snext = (size_t)((s + 1) & (S_LEN - 1)) * B_LEN;
    float xn[NCH];
#pragma unroll
    for (int ch = 0; ch < NCH; ++ch) xn[ch] = xp[ch][snext];

    float outv[NCH];
#pragma unroll
    for (int ch = 0; ch < NCH; ++ch) {
      // ---- layer 0: G0^T = W_hh0p * h0^T + (x*W_ih0p + b0p) --------------
      v8f C0;
#pragma unroll
      for (int r = 0; r < 8; ++r) C0[r] = xv[ch] * w0g[r] + bs0[r];
      v2f B0;  B0.x = h0a[ch];  B0.y = h0b[ch];  // already in B layout!
      v8f D0 = __builtin_amdgcn_wmma_f32_16x16x4_f32(
          false, A0, false, B0, (short)0, C0, false, false);
      cell4(D0[0], D0[1], D0[2], D0[3], c0a[ch], h0a[ch]);
      cell4(D0[4], D0[5], D0[6], D0[7], c0b[ch], h0b[ch]);

      // ---- layer 1: G1^T = W_ih1p*h0^T + W_hh1p*h1^T + b1p ---------------
      v2f Bx;  Bx.x = h0a[ch];  Bx.y = h0b[ch];
      v8f D1 = __builtin_amdgcn_wmma_f32_16x16x4_f32(
          false, A1i, false, Bx, (short)0, bias1, false, false);
      v2f Bh;  Bh.x = h1a[ch];  Bh.y = h1b[ch];
      D1 = __builtin_amdgcn_wmma_f32_16x16x4_f32(
          false, A1h, false, Bh, (short)0, D1, false, false);
      cell4(D1[0], D1[1], D1[2], D1[3], c1a[ch], h1a[ch]);
      cell4(D1[4], D1[5], D1[6], D1[7], c1b[ch], h1b[ch]);

      // ---- output: dot(W_out, h1) needs lo+hi halves -> one SWAPX16 ------
      float part = woA * h1a[ch] + woB * h1b[ch];
      float swp  = __int_as_float(
          __builtin_amdgcn_ds_swizzle(__float_as_int(part), 0x401f));
      outv[ch] = part + swp + bo;
      xv[ch]   = xn[ch];
    }

    if (!hi) {  // lo/hi halves hold identical sums; store once
      const size_t so = (size_t)s * B_LEN;
#pragma unroll
      for (int ch = 0; ch < NCH; ++ch) op[ch][so] = outv[ch];
    }
  }
}

extern "C" void kernel_launch(void* const* d_in, const int* in_sizes, int n_in,
                              void* d_out, int out_size, void* d_ws,
                              size_t ws_size, hipStream_t stream) {
  (void)in_sizes; (void)n_in; (void)out_size; (void)d_ws; (void)ws_size;
  dim3 grid(B_LEN / (16 * NCH)), block(32);  // 128 waves x 2 chains
  lstm2_wmma_kernel<<<grid, block, 0, stream>>>(
      (const float*)d_in[0],                          // x
      (const float*)d_in[1], (const float*)d_in[2],   // W_ih0, W_hh0
      (const float*)d_in[3], (const float*)d_in[4],   // b_ih0, b_hh0
      (const float*)d_in[5], (const float*)d_in[6],   // W_ih1, W_hh1
      (const float*)d_in[7], (const float*)d_in[8],   // b_ih1, b_hh1
      (const float*)d_in[9], (const float*)d_in[10],  // W_out, b_out
      (float*)d_out);
}